// RandomLowRes2D_76003741270470
// MI455X (gfx1250) — compile-verified
//
#include <hip/hip_runtime.h>
#include <math.h>

#define RLR_H    512
#define RLR_TC   32          // columns (non-blur axis) per tile
#define RLR_STR  33          // padded LDS row stride (gcd(33,64)=1 -> conflict-free)
#define RLR_R    15          // Gaussian truncation radius (matches reference)
#define RLR_NT   256         // threads per block = 8 waves (wave32)
#define RLR_LDS_FLOATS (2 * RLR_H * RLR_STR + 32)

#if defined(__has_builtin)
#  if __has_builtin(__builtin_amdgcn_global_load_async_to_lds_b32)
#    define RLR_HAVE_ASYNC 1
#  endif
#  if __has_builtin(__builtin_amdgcn_s_wait_asynccnt)
#    define RLR_HAVE_WAITASYNC 1
#  endif
#endif

#if defined(RLR_HAVE_ASYNC)
typedef __attribute__((address_space(1))) int rlr_gint;   // global (AS1)
typedef __attribute__((address_space(3))) int rlr_lint;   // LDS (AS3)
#endif

__device__ __forceinline__ void rlr_wait_async0() {
#if defined(RLR_HAVE_WAITASYNC)
    __builtin_amdgcn_s_wait_asynccnt(0);
#elif defined(RLR_HAVE_ASYNC)
    asm volatile("s_wait_asynccnt 0" ::: "memory");
#endif
}

// --------------------------------------------------------------------------
// Probe kernel (never launched): placed FIRST in the file so the disasm
// snippet shows whether global_load_async_to_lds_b32 / s_wait_asynccnt lower.
// --------------------------------------------------------------------------
extern "C" __global__ void rlr_async_probe(float* g) {
#if defined(RLR_HAVE_ASYNC)
    __shared__ float buf[32];
    __builtin_amdgcn_global_load_async_to_lds_b32(
        (rlr_gint*)(g + threadIdx.x),
        (rlr_lint*)(buf + threadIdx.x), 0, 0);
    rlr_wait_async0();
    __syncthreads();
    g[threadIdx.x] = buf[threadIdx.x] + 1.0f;   // keep the copy alive
#else
    g[threadIdx.x] = 0.0f;
#endif
}

// --------------------------------------------------------------------------
// Main kernel. One workgroup handles one (image, 32-wide strip) tile.
// u = coordinate along the blur axis (0..511), v = strip coordinate (0..31).
// LDS tile is always [u][v] so the blur/resample code is axis-agnostic;
// only the global<->(u,v) address mapping changes with `ax`.
// --------------------------------------------------------------------------
extern "C" __global__ __launch_bounds__(RLR_NT)
void rlr2d_kernel(const float* __restrict__ x,
                  const float* __restrict__ resolution,
                  const int*   __restrict__ axis,
                  const float* __restrict__ gapv,
                  float*       __restrict__ out)
{
    extern __shared__ float lds[];
    float* raw = lds;                              // [512][33] staged input tile
    float* low = lds + RLR_H * RLR_STR;            // [512][33] blurred+downsampled rows
    float* wl  = lds + 2 * RLR_H * RLR_STR;        // [31] unnormalized Gaussian weights

    const int n   = blockIdx.y;                    // image index (B*C)
    const int c0  = blockIdx.x * RLR_TC;           // strip origin on the non-blur axis
    const int tid = threadIdx.x;

    const float res = resolution[n];
    const int   ax  = axis[n];
    const float gp  = gapv[n];
    // sigma = fwhm / sqrt(8 ln 2), fwhm = res * gap
    const float sigma = fmaxf(res * gp * 0.42466090014400953f, 1e-6f);

    const size_t base = (size_t)n * (RLR_H * RLR_H);
    const float* img  = x + base;

    // ---- Phase 1: stage tile into LDS via async global->LDS ----
#pragma unroll 4
    for (int e = tid; e < RLR_H * RLR_TC; e += RLR_NT) {
        int u, v, gidx;
        if (ax == 0) { u = e >> 5;          v = e & 31; gidx = (u << 9) + (c0 + v); }
        else         { u = e & (RLR_H - 1); v = e >> 9; gidx = ((c0 + v) << 9) + u; }
        float* dst = raw + u * RLR_STR + v;
#if defined(RLR_HAVE_ASYNC)
        __builtin_amdgcn_global_load_async_to_lds_b32(
            (rlr_gint*)(img + gidx),
            (rlr_lint*)dst,
            /*offset=*/0, /*cpol=*/0);
#else
        *dst = img[gidx];
#endif
    }

    // Overlap weight computation with the async copy.
    if (tid < 2 * RLR_R + 1) {
        float o = (float)(tid - RLR_R);
        float t = o / sigma;
        wl[tid] = __expf(-0.5f * t * t);
    }

    rlr_wait_async0();
    __syncthreads();

    float wsum = 0.0f;
#pragma unroll
    for (int k = 0; k < 2 * RLR_R + 1; ++k) wsum += wl[k];
    const float invw = 1.0f / wsum;

    int n_low = (int)floorf((float)RLR_H / res);
    if (n_low < 1) n_low = 1;
    // Taps beyond ~7.5*sigma are < 1e-12 of the center weight: skip them.
    const int kr = (int)fminf(ceilf(7.5f * sigma), (float)RLR_R);

    const int v  = tid & 31;   // lane -> strip column (consecutive -> no bank conflicts)
    const int rr = tid >> 5;   // wave-row group

    // ---- Phase 2: fused blur + downsample into low[0..n_low) ----
    // low[i] = (1-fr)*blur(lo) + fr*blur(lo+1) == single (2kr+2)-tap filter.
    for (int i = rr; i < n_low; i += RLR_NT / 32) {
        float pos = fminf((float)i * res, (float)(RLR_H - 1));
        float lof = floorf(pos);
        float fr  = pos - lof;
        int   lo  = (int)lof;
        if (lo >= RLR_H - 1) { lo = RLR_H - 1; fr = 0.0f; }   // hi==lo edge

        float acc = 0.0f;
        for (int k = -kr; k <= kr + 1; ++k) {
            int r = lo + k;                               // symmetric padding
            r = (r < 0) ? (-1 - r) : r;
            r = (r > RLR_H - 1) ? (2 * RLR_H - 1 - r) : r;
            float w0 = (k <= RLR_R && k >= -RLR_R) ? wl[k + RLR_R]     : 0.0f;
            float w1 = (k - 1 >= -RLR_R)           ? wl[k - 1 + RLR_R] : 0.0f;
            float cw = (1.0f - fr) * w0 + fr * w1;
            acc += cw * raw[r * RLR_STR + v];
        }
        low[i * RLR_STR + v] = acc * invw;
    }

    __syncthreads();

    // ---- Phase 3: upsample from low, coalesced store ----
    float* outp = out + base;
    const float nl1 = (float)(n_low - 1);
#pragma unroll 2
    for (int e = tid; e < RLR_H * RLR_TC; e += RLR_NT) {
        int u, vv, gidx;
        if (ax == 0) { u = e >> 5;          vv = e & 31; gidx = (u << 9) + (c0 + vv); }
        else         { u = e & (RLR_H - 1); vv = e >> 9; gidx = ((c0 + vv) << 9) + u; }
        float pos2 = fminf((float)u / res, nl1);
        float lof  = floorf(pos2);
        float fr2  = pos2 - lof;
        int lo2 = (int)lof; if (lo2 > n_low - 1) lo2 = n_low - 1;
        int hi2 = lo2 + 1;  if (hi2 > n_low - 1) hi2 = n_low - 1;
        float a = low[lo2 * RLR_STR + vv];
        float b = low[hi2 * RLR_STR + vv];
        outp[gidx] = a * (1.0f - fr2) + b * fr2;
    }
}

extern "C" void kernel_launch(void* const* d_in, const int* in_sizes, int n_in,
                              void* d_out, int out_size, void* d_ws, size_t ws_size,
                              hipStream_t stream) {
    (void)n_in; (void)out_size; (void)d_ws; (void)ws_size;
    const float* x          = (const float*)d_in[0];
    const float* resolution = (const float*)d_in[1];
    const int*   axis       = (const int*)d_in[2];
    const float* gap        = (const float*)d_in[3];
    float*       out        = (float*)d_out;

    const int nimg  = in_sizes[1];                 // B*C = 64
    const size_t sh = RLR_LDS_FLOATS * sizeof(float);   // ~132 KB of LDS

    // Allow >64KB dynamic LDS (deterministic, idempotent; not a capture-breaking op).
    (void)hipFuncSetAttribute(reinterpret_cast<const void*>(rlr2d_kernel),
                              hipFuncAttributeMaxDynamicSharedMemorySize, (int)sh);

    dim3 grid(RLR_H / RLR_TC, nimg);               // 16 strips x 64 images
    rlr2d_kernel<<<grid, RLR_NT, sh, stream>>>(x, resolution, axis, gap, out);
}